// StructuredLogits_46875273069167
// MI455X (gfx1250) — compile-verified
//
#include <hip/hip_runtime.h>
#include <hip/hip_bf16.h>
#include <stdint.h>

// ---------------------------------------------------------------------------
// StructuredLogits forward:  out[m,i] = (sum_e{rows[e]==i} vals[e]*logits[m,cols[e]])
//                                       / rowsum[i] + logits[m,i]
// M = 128 (B*S), N = 30000, NNZ = 480000.  Pure data-movement problem
// (~123 MFLOP vs ~300 MB of L2-resident traffic) -> WMMA has no valid tile
// structure here (per-row private column indices); the win is layout + CSR.
//
// Pipeline:
//   1) transpose logits [128,N] -> flat_t [N,128]   (async g->LDS, LDS->g)
//   2) counting sort COO -> CSR (int atomics + single-block scan)
//   3) one wave32 per row: register-accumulate edges, fuse /rowsum and
//      +logits residual (flat_t[i,:] IS logits[:,i]), async LDS->global
//      transposed write-out.
// ---------------------------------------------------------------------------

#define AS1 __attribute__((address_space(1)))
#define AS3 __attribute__((address_space(3)))

__device__ __forceinline__ void async_g2l_b32(const void* g, void* l) {
  __builtin_amdgcn_global_load_async_to_lds_b32((AS1 int*)(AS1 void*)g,
                                                (AS3 int*)(AS3 void*)l, 0, 0);
}
__device__ __forceinline__ void async_l2g_b32(void* g, const void* l) {
  __builtin_amdgcn_global_store_async_from_lds_b32((AS1 int*)(AS1 void*)g,
                                                   (AS3 int*)(AS3 void*)l, 0, 0);
}
__device__ __forceinline__ void wait_async() {
  __builtin_amdgcn_s_wait_asynccnt(0);
}

// ---------------- kernel 1: zero the per-row counters ------------------------
__global__ __launch_bounds__(256) void k_zero_i32(int* __restrict__ p, int n) {
  int i = blockIdx.x * blockDim.x + threadIdx.x;
  if (i < n) p[i] = 0;
}

// ---------------- kernel 2: logits [M,N] -> flat_t [N,M] ---------------------
// 32x32 tile, 8 wave32.  Phase A: async global->LDS with transposed per-lane
// scatter.  Phase B: async LDS->global stores.  Both coalesced on the global
// side; LDS pad 33 keeps banks clean.
__global__ __launch_bounds__(256) void k_transpose_in(const float* __restrict__ logits,
                                                      float* __restrict__ flat_t,
                                                      int N, int M) {
  __shared__ float tile[32][33];            // [n_local][m_local]
  const int tx = threadIdx.x & 31;
  const int ty = threadIdx.x >> 5;          // 0..7
  const int n0 = blockIdx.x * 32;
  const int m0 = blockIdx.y * 32;
  const int n  = n0 + tx;

  if (n < N) {
#pragma unroll
    for (int r = 0; r < 4; ++r) {
      const int ml = ty + 8 * r;
      async_g2l_b32(logits + (size_t)(m0 + ml) * N + n, &tile[tx][ml]);
    }
  }
  wait_async();
  __syncthreads();

#pragma unroll
  for (int r = 0; r < 4; ++r) {
    const int nl = ty + 8 * r;
    if (n0 + nl < N)
      async_l2g_b32(flat_t + (size_t)(n0 + nl) * M + m0 + tx, &tile[nl][tx]);
  }
  wait_async();
}

// ---------------- kernel 3: per-row edge counts ------------------------------
__global__ __launch_bounds__(256) void k_count(const int* __restrict__ rows,
                                               int* __restrict__ off, int nnz) {
  int e = blockIdx.x * blockDim.x + threadIdx.x;
  if (e < nnz)
    __hip_atomic_fetch_add(&off[rows[e]], 1, __ATOMIC_RELAXED, __HIP_MEMORY_SCOPE_AGENT);
}

// ---------------- kernel 4: single-block exclusive scan over off[0..n) -------
// 30 chunks of 1024; Hillis-Steele in LDS with a running carry.
__global__ __launch_bounds__(1024) void k_scan(int* __restrict__ off, int n) {
  __shared__ int sdata[1024];
  __shared__ int carry_s;
  const int t = threadIdx.x;
  if (t == 0) carry_s = 0;
  __syncthreads();
  for (int base = 0; base < n; base += 1024) {
    const int idx = base + t;
    const int x = (idx < n) ? off[idx] : 0;
    sdata[t] = x;
    __syncthreads();
    for (int d = 1; d < 1024; d <<= 1) {
      int y = (t >= d) ? sdata[t - d] : 0;
      __syncthreads();
      sdata[t] += y;
      __syncthreads();
    }
    const int incl  = sdata[t];
    const int carry = carry_s;
    if (idx < n) off[idx] = carry + incl - x;   // exclusive prefix
    __syncthreads();                            // everyone read carry_s
    if (t == 0) carry_s = carry + sdata[1023];
    __syncthreads();
  }
}

// ---------------- kernel 5: scatter edges into CSR order ---------------------
// fetch_add turns off[i] into the END of row i; row i = [off[i-1], off[i]).
__global__ __launch_bounds__(256) void k_fill(const int* __restrict__ rows,
                                              const int* __restrict__ cols,
                                              const float* __restrict__ vals,
                                              int* __restrict__ off,
                                              int* __restrict__ ccols,
                                              float* __restrict__ cvals, int nnz) {
  int e = blockIdx.x * blockDim.x + threadIdx.x;
  if (e >= nnz) return;
  int pos = __hip_atomic_fetch_add(&off[rows[e]], 1,
                                   __ATOMIC_RELAXED, __HIP_MEMORY_SCOPE_AGENT);
  ccols[pos] = cols[e];
  cvals[pos] = vals[e];
}

// ---------------- kernel 6 (hot): fused gather + normalize + residual --------
// 32 wave32 per block; wave w owns row i = i0 + w.  Lane l holds float4 of the
// 128-long vector.  Register accumulation (no float atomics), rowsum computed
// in the same loop, residual comes from flat_t[i,:] (== logits[:,i]).
// Finished rows staged in LDS, written out transposed via async LDS->global.
#define OPITCH 132   // floats; 132*4=528 bytes, 16B-aligned rows for b128 LDS stores
__global__ __launch_bounds__(1024) void k_gather(const float* __restrict__ flat_t,
                                                 const int* __restrict__ off,
                                                 const int* __restrict__ ccols,
                                                 const float* __restrict__ cvals,
                                                 float* __restrict__ out, int N, int M) {
  __shared__ float tile[32 * OPITCH];       // [i_local][m]
  const int lane = threadIdx.x & 31;
  const int w    = threadIdx.x >> 5;        // 0..31 = i_local
  const int i0   = blockIdx.x * 32;
  const int i    = i0 + w;

  int start = 0, end = 0;
  if (i < N) {                              // wave-uniform scalar loads
    start = (i > 0) ? off[i - 1] : 0;
    end   = off[i];
  }

  float4 a = make_float4(0.f, 0.f, 0.f, 0.f);
  float  rs = 0.f;
  for (int e = start; e < end; ++e) {
    const int   c = ccols[e];               // uniform -> s_load
    const float v = cvals[e];
    const float4 x = reinterpret_cast<const float4*>(flat_t + (size_t)c * 128)[lane];
    a.x += v * x.x;  a.y += v * x.y;  a.z += v * x.z;  a.w += v * x.w;
    rs  += v;
  }

  if (i < N) {
    const float  inv  = 1.0f / rs;          // every row has >=1 edge (reference)
    const float4 self = reinterpret_cast<const float4*>(flat_t + (size_t)i * 128)[lane];
    float4 res;
    res.x = a.x * inv + self.x;
    res.y = a.y * inv + self.y;
    res.z = a.z * inv + self.z;
    res.w = a.w * inv + self.w;
    reinterpret_cast<float4*>(&tile[w * OPITCH])[lane] = res;   // ds_store_b128
  }
  __syncthreads();

  // transposed write-out: lanes sweep i (coalesced 128B lines), 4 m per thread
  const int il = threadIdx.x & 31;
  const int mb = threadIdx.x >> 5;          // 0..31
  if (i0 + il < N) {
#pragma unroll
    for (int r = 0; r < 4; ++r) {
      const int m = mb + 32 * r;            // 0..127
      async_l2g_b32(out + (size_t)m * N + i0 + il, &tile[il * OPITCH + m]);
    }
  }
  wait_async();
}

// ---------------------------------------------------------------------------
extern "C" void kernel_launch(void* const* d_in, const int* in_sizes, int n_in,
                              void* d_out, int out_size, void* d_ws, size_t ws_size,
                              hipStream_t stream) {
  const float* logits = (const float*)d_in[0];
  const int*   adj    = (const int*)d_in[1];
  const float* vals   = (const float*)d_in[2];

  const int M   = 128;                      // B*S from the reference
  const int N   = in_sizes[0] / M;          // 30000
  const int NNZ = in_sizes[2];              // 480000
  const int* rows = adj;
  const int* cols = adj + NNZ;

  // workspace: [flat_t N*M f32][off N i32][ccols NNZ i32][cvals NNZ f32] ~19.3MB
  float* flat_t = (float*)d_ws;
  int*   off    = (int*)(flat_t + (size_t)N * M);
  int*   ccols  = off + N;
  float* cvals  = (float*)(ccols + NNZ);

  k_zero_i32<<<(N + 255) / 256, 256, 0, stream>>>(off, N);

  dim3 tgrid((N + 31) / 32, M / 32);
  k_transpose_in<<<tgrid, 256, 0, stream>>>(logits, flat_t, N, M);

  k_count<<<(NNZ + 255) / 256, 256, 0, stream>>>(rows, off, NNZ);
  k_scan<<<1, 1024, 0, stream>>>(off, N);
  k_fill<<<(NNZ + 255) / 256, 256, 0, stream>>>(rows, cols, vals, off, ccols, cvals, NNZ);

  k_gather<<<(N + 31) / 32, 1024, 0, stream>>>(flat_t, off, ccols, cvals,
                                               (float*)d_out, N, M);
}